// HybridGeometryFeatures_83915071030245
// MI455X (gfx1250) — compile-verified
//
#include <hip/hip_runtime.h>
#include <math.h>

// ---------------------------------------------------------------------------
// HybridGeometryFeatures for MI455X (gfx1250, wave32)
//
// KNN (K=16) over N=12288 points per batch via fused 16x16 distance tiles:
//   dist2 = |q|^2 + |c|^2 - 2 * (Q . C^T)
// The 16x16 Gram tile is computed with V_WMMA_F32_16X16X4_F32 (xyz padded to
// K=4 with 0), full fp32 precision. Top-16 lists are kept in LDS and updated
// on the fly (no NxN matrix is ever materialized). Features (centroid, 3x3
// covariance, closed-form symmetric eigendecomposition, PCA normal) are
// computed in the same kernel by lanes 0-15 of each wave.
//
// All LDS traffic is wave-internal (per-wave disjoint slices). CDNA5 keeps
// same-wave DS instructions in order, so instead of block-wide s_barrier
// pairs per tile we use a compiler memory barrier + s_wait_dscnt 0 — the 8
// waves of a block run fully decoupled.
// ---------------------------------------------------------------------------

#define NPTS 12288
#define NB   2
#define KNN  16
#define WPB  8          // waves per block
#define TPB  (NPTS/16)  // candidate tiles per batch (768)

typedef __attribute__((ext_vector_type(2))) float v2f;
typedef __attribute__((ext_vector_type(8))) float v8f;

// Wave-internal LDS fence: compiler scheduling barrier + wait for DS ops.
// Same-wave DS ops are in-order on CDNA5, so this is sufficient for
// cross-lane store->load visibility within one wave.
__device__ __forceinline__ void wave_lds_fence() {
    asm volatile("s_wait_dscnt 0x0" ::: "memory");
}

__global__ __launch_bounds__(256)
void HybridGeometryFeatures_knn_wmma_kernel(const float* __restrict__ xyz,
                                            const float* __restrict__ nrm,
                                            float* __restrict__ out)
{
    __shared__ float ldsQ [WPB][16];        // |query|^2 per row
    __shared__ float ldsD [WPB][256];       // 16x16 distance tile per wave
    __shared__ float ldsKd[WPB][16][KNN];   // top-K distances
    __shared__ int   ldsKi[WPB][16][KNN];   // top-K indices

    const int lane = threadIdx.x & 31;
    const int wave = threadIdx.x >> 5;
    const int row  = lane & 15;             // N column of C tile / query row id
    const int half = lane >> 4;             // 0: lanes 0-15, 1: lanes 16-31

    const int tile_global = blockIdx.x * WPB + wave;
    const int b      = tile_global / TPB;
    const int tile_m = tile_global % TPB;
    const int qbase  = tile_m * 16;

    const float* P   = xyz + (size_t)b * NPTS * 3;
    const float* Nrm = nrm + (size_t)b * NPTS * 3;

    // ---- A operand: 16x4 f32 layout (lanes 0-15: K={0,1}; 16-31: K={2,3}) --
    const int q = qbase + row;
    const float qx = P[q*3+0], qy = P[q*3+1], qz = P[q*3+2];
    v2f Aop;
    Aop.x = half ? qz : qx;
    Aop.y = half ? 0.0f : qy;

    if (half == 0) {
        ldsQ[wave][row] = qx*qx + qy*qy + qz*qz;
        #pragma unroll
        for (int t = 0; t < KNN; ++t) {
            ldsKd[wave][row][t] = __builtin_inff();
            ldsKi[wave][row][t] = 0;
        }
    }
    wave_lds_fence();

    // per-lane loop-invariant query norms for its 8 C rows (M = half*8 + v)
    float qsqv[8];
    #pragma unroll
    for (int v = 0; v < 8; ++v) qsqv[v] = ldsQ[wave][half*8 + v];

    float kmax = __builtin_inff();
    int   karg = 0;

    // ---- sweep all candidate tiles -----------------------------------------
    for (int j = 0; j < TPB; ++j) {
        const int c = j*16 + row;
        const float cx = P[c*3+0], cy = P[c*3+1], cz = P[c*3+2];
        if (j + 1 < TPB)  // next tile is L2-resident; pull it toward the WGP
            __builtin_prefetch(&P[(c + 16) * 3], 0, 1);
        const float csq = cx*cx + cy*cy + cz*cz;

        // B operand: 4x16 f32 (lanes 0-15: K={0,1}; lanes 16-31: K={2,3})
        v2f Bop;
        Bop.x = half ? cz : cx;
        Bop.y = half ? 0.0f : cy;

        v8f C = {};
        C = __builtin_amdgcn_wmma_f32_16x16x4_f32(
                /*neg_a=*/false, Aop, /*neg_b=*/false, Bop,
                /*c_mod=*/(short)0, C, /*reuse_a=*/false, /*reuse_b=*/false);

        // lane holds C rows M = half*8+v, column N = row
        #pragma unroll
        for (int v = 0; v < 8; ++v) {
            const int M = half*8 + v;
            float d = qsqv[v] + csq - 2.0f * C[v];
            d = fmaxf(d, 0.0f);
            if (j*16 + row == qbase + M) d = __builtin_inff();  // mask self
            ldsD[wave][M*16 + row] = d;
        }
        wave_lds_fence();

        // lanes 0-15: fold the 16 candidates of this tile into row's top-K
        if (half == 0) {
            for (int n = 0; n < 16; ++n) {
                const float d = ldsD[wave][row*16 + n];
                if (d < kmax) {
                    ldsKd[wave][row][karg] = d;
                    ldsKi[wave][row][karg] = j*16 + n;
                    float m = -__builtin_inff(); int a = 0;
                    #pragma unroll
                    for (int t = 0; t < KNN; ++t) {
                        const float dt = ldsKd[wave][row][t];
                        if (dt > m) { m = dt; a = t; }
                    }
                    kmax = m; karg = a;
                }
            }
        }
        wave_lds_fence();
    }

    // ---- features: lanes 0-15, one query each ------------------------------
    if (half == 0) {
        int idxs[KNN];
        #pragma unroll
        for (int t = 0; t < KNN; ++t) idxs[t] = ldsKi[wave][row][t];

        // pass 1: centroid
        float mx = 0.f, my = 0.f, mz = 0.f;
        #pragma unroll
        for (int t = 0; t < KNN; ++t) {
            const int nb = idxs[t];
            mx += P[nb*3+0]; my += P[nb*3+1]; mz += P[nb*3+2];
        }
        const float invK = 1.0f / (float)KNN;
        mx *= invK; my *= invK; mz *= invK;

        // pass 2: covariance (6 unique terms), / (K-1)
        float sxx=0.f, sxy=0.f, sxz=0.f, syy=0.f, syz=0.f, szz=0.f;
        #pragma unroll
        for (int t = 0; t < KNN; ++t) {
            const int nb = idxs[t];
            const float dx = P[nb*3+0] - mx;
            const float dy = P[nb*3+1] - my;
            const float dz = P[nb*3+2] - mz;
            sxx += dx*dx; sxy += dx*dy; sxz += dx*dz;
            syy += dy*dy; syz += dy*dz; szz += dz*dz;
        }
        const float invKm1 = 1.0f / (float)(KNN - 1);
        const float a00 = sxx*invKm1, a11 = syy*invKm1, a22 = szz*invKm1;
        const float a01 = sxy*invKm1, a02 = sxz*invKm1, a12 = syz*invKm1;

        // symmetric 3x3 eigenvalues, Smith's trigonometric method
        const float p1 = a01*a01 + a02*a02 + a12*a12;
        const float qm = (a00 + a11 + a22) * (1.0f/3.0f);
        const float d0 = a00 - qm, d1 = a11 - qm, d2 = a22 - qm;
        const float p2 = d0*d0 + d1*d1 + d2*d2 + 2.0f*p1;
        const float p  = sqrtf(fmaxf(p2 * (1.0f/6.0f), 1e-30f));
        const float ip = 1.0f / p;
        const float b00 = d0*ip, b11 = d1*ip, b22 = d2*ip;
        const float b01 = a01*ip, b02 = a02*ip, b12 = a12*ip;
        const float detB = b00*(b11*b22 - b12*b12)
                         - b01*(b01*b22 - b12*b02)
                         + b02*(b01*b12 - b11*b02);
        const float rr   = fminf(fmaxf(0.5f*detB, -1.0f), 1.0f);
        const float phi  = acosf(rr) * (1.0f/3.0f);
        const float e1 = qm + 2.0f*p*cosf(phi);                       // largest
        const float e3 = qm + 2.0f*p*cosf(phi + 2.0943951023931953f); // smallest
        const float e2 = 3.0f*qm - e1 - e3;

        // eigenvector of smallest eigenvalue: cross products of rows of A-e3*I
        const float r0x = a00 - e3, r0y = a01,      r0z = a02;
        const float r1x = a01,      r1y = a11 - e3, r1z = a12;
        const float r2x = a02,      r2y = a12,      r2z = a22 - e3;
        float c0x = r0y*r1z - r0z*r1y, c0y = r0z*r1x - r0x*r1z, c0z = r0x*r1y - r0y*r1x;
        float c1x = r0y*r2z - r0z*r2y, c1y = r0z*r2x - r0x*r2z, c1z = r0x*r2y - r0y*r2x;
        float c2x = r1y*r2z - r1z*r2y, c2y = r1z*r2x - r1x*r2z, c2z = r1x*r2y - r1y*r2x;
        const float n0 = c0x*c0x + c0y*c0y + c0z*c0z;
        const float n1 = c1x*c1x + c1y*c1y + c1z*c1z;
        const float n2 = c2x*c2x + c2y*c2y + c2z*c2z;
        float vx = c0x, vy = c0y, vz = c0z, nn = n0;
        if (n1 > nn) { vx = c1x; vy = c1y; vz = c1z; nn = n1; }
        if (n2 > nn) { vx = c2x; vy = c2y; vz = c2z; nn = n2; }
        if (nn > 1e-30f) {
            const float inv = rsqrtf(nn);
            vx *= inv; vy *= inv; vz *= inv;
        } else {
            vx = 0.f; vy = 0.f; vz = 1.f;
        }

        // consistency / sign-fix
        const float nx0 = Nrm[q*3+0], ny0 = Nrm[q*3+1], nz0 = Nrm[q*3+2];
        const float dotn = vx*nx0 + vy*ny0 + vz*nz0;
        const float sgn = (dotn > 0.f) ? 1.f : ((dotn < 0.f) ? -1.f : 0.f);
        vx *= sgn; vy *= sgn; vz *= sgn;
        const float consistency = fabsf(dotn);

        const float tot = fmaxf(e1 + e2 + e3, 1e-8f);
        const float curvature = e3 / tot;

        // pass 3: roughness = mean |centered . pca_normal|
        float rough = 0.f;
        #pragma unroll
        for (int t = 0; t < KNN; ++t) {
            const int nb = idxs[t];
            const float dx = P[nb*3+0] - mx;
            const float dy = P[nb*3+1] - my;
            const float dz = P[nb*3+2] - mz;
            rough += fabsf(dx*vx + dy*vy + dz*vz);
        }
        rough *= invK;

        float* o = out + ((size_t)b * NPTS + q) * 6;
        o[0] = consistency;
        o[1] = nx0; o[2] = ny0; o[3] = nz0;
        o[4] = curvature;
        o[5] = rough;
    }
}

extern "C" void kernel_launch(void* const* d_in, const int* in_sizes, int n_in,
                              void* d_out, int out_size, void* d_ws, size_t ws_size,
                              hipStream_t stream) {
    (void)in_sizes; (void)n_in; (void)d_ws; (void)ws_size; (void)out_size;
    const float* xyz = (const float*)d_in[0];
    const float* nrm = (const float*)d_in[1];
    float* out = (float*)d_out;
    // 2 batches * 12288 queries / (8 waves * 16 queries per wave) = 192 blocks
    const int blocks = (NB * NPTS) / (WPB * 16);
    HybridGeometryFeatures_knn_wmma_kernel<<<blocks, 256, 0, stream>>>(xyz, nrm, out);
}